// RISController_39367670235722
// MI455X (gfx1250) — compile-verified
//
#include <hip/hip_runtime.h>

typedef __attribute__((ext_vector_type(2))) float v2f;
typedef __attribute__((ext_vector_type(8))) float v8f;

#define T_STEPS 16
#define BETA 0.9f
#define THR  1.0f

// One wave (32 lanes) processes a 16-row batch tile.
// A-layout (16x4 f32): lane<16 holds K={4k,4k+1} of row M=lane; lane>=16 holds K={4k+2,4k+3} of row M=lane-16.
// C-layout (16x16 f32): lane<16 holds (M=v, N=lane); lane>=16 holds (M=v+8, N=lane-16).
__global__ __launch_bounds__(256) void snn_lif_wmma_kernel(
    const float* __restrict__ x,     // [B,3]
    const float* __restrict__ W1,    // [32,3]
    const float* __restrict__ b1,    // [32]
    const float* __restrict__ Wout,  // [9,4,32] -> flat [36,32]
    const float* __restrict__ bout,  // [9,4]    -> flat [36]
    float* __restrict__ out,         // [B,36]
    int B)
{
    const int lane    = threadIdx.x & 31;
    const int wave    = threadIdx.x >> 5;
    const int rowBase = (blockIdx.x * 8 + wave) * 16;
    if (rowBase >= B) return;                 // wave-uniform exit: EXEC stays all-ones

    const int halfSel = lane >> 4;            // 0: lanes 0-15, 1: lanes 16-31
    const int mlane   = lane & 15;
    const int hbase   = halfSel * 2;          // K offset inside each group of 4

    // ---- load this lane's x row (A-side row) ----
    int arow = rowBase + mlane;
    if (arow >= B) arow = B - 1;              // safety clamp (B divides evenly in practice)
    const float x0 = x[arow * 3 + 0];
    const float x1 = x[arow * 3 + 1];
    const float x2 = x[arow * 3 + 2];

    // ---- in1 = x @ W1^T + b1, in A-fragment layout (constant over timesteps) ----
    v2f in1[8];
#pragma unroll
    for (int k4 = 0; k4 < 8; ++k4) {
        const int h0 = 4 * k4 + hbase;
        const int h1 = h0 + 1;
        in1[k4].x = x0 * W1[h0 * 3 + 0] + x1 * W1[h0 * 3 + 1] + x2 * W1[h0 * 3 + 2] + b1[h0];
        in1[k4].y = x0 * W1[h1 * 3 + 0] + x1 * W1[h1 * 3 + 1] + x2 * W1[h1 * 3 + 2] + b1[h1];
    }

    // ---- B fragments of Wout (K x N = 4 x 16 tiles; 3 N-tiles, 8 K-groups), zero-padded N>=36 ----
    v2f   bw[3][8];
    float boutS[3];
#pragma unroll
    for (int nt = 0; nt < 3; ++nt) {
        const int  col   = 16 * nt + mlane;   // flattened head index j = n*4+o
        const bool valid = (col < 36);
        boutS[nt] = valid ? bout[col] : 0.0f;
#pragma unroll
        for (int k4 = 0; k4 < 8; ++k4) {
            const int k0 = 4 * k4 + hbase;
            bw[nt][k4].x = valid ? Wout[col * 32 + k0]     : 0.0f;
            bw[nt][k4].y = valid ? Wout[col * 32 + k0 + 1] : 0.0f;
        }
    }

    // ---- state ----
    v2f mem1[8];
#pragma unroll
    for (int k4 = 0; k4 < 8; ++k4) { mem1[k4].x = 0.0f; mem1[k4].y = 0.0f; }

    v8f memo[3], counts[3];
#pragma unroll
    for (int nt = 0; nt < 3; ++nt) {
        memo[nt]   = (v8f){0.f, 0.f, 0.f, 0.f, 0.f, 0.f, 0.f, 0.f};
        counts[nt] = (v8f){0.f, 0.f, 0.f, 0.f, 0.f, 0.f, 0.f, 0.f};
    }

    v2f aspk[8];

    for (int t = 0; t < T_STEPS; ++t) {
        // layer-1 LIF: reset from previous mem, integrate, spike (elementwise in A layout)
#pragma unroll
        for (int k4 = 0; k4 < 8; ++k4) {
            const float rx = (mem1[k4].x > THR) ? THR : 0.0f;
            const float ry = (mem1[k4].y > THR) ? THR : 0.0f;
            mem1[k4].x = BETA * mem1[k4].x + in1[k4].x - rx;
            mem1[k4].y = BETA * mem1[k4].y + in1[k4].y - ry;
            aspk[k4].x = (mem1[k4].x > THR) ? 1.0f : 0.0f;
            aspk[k4].y = (mem1[k4].y > THR) ? 1.0f : 0.0f;
        }

        // output heads: ino = spk1 @ Wout^T + bout via exact fp32 WMMA (K=32 as 8 x K4)
#pragma unroll
        for (int nt = 0; nt < 3; ++nt) {
            v8f acc;
#pragma unroll
            for (int v = 0; v < 8; ++v) acc[v] = boutS[nt];   // bout broadcast down M
#pragma unroll
            for (int k4 = 0; k4 < 8; ++k4) {
                acc = __builtin_amdgcn_wmma_f32_16x16x4_f32(
                    false, aspk[k4], false, bw[nt][k4], (short)0, acc, false, false);
            }
            // output LIF + spike counting (elementwise in C layout)
#pragma unroll
            for (int v = 0; v < 8; ++v) {
                float m = memo[nt][v];
                const float r = (m > THR) ? THR : 0.0f;
                m = BETA * m + acc[v] - r;
                memo[nt][v] = m;
                counts[nt][v] += (m > THR) ? 1.0f : 0.0f;
            }
        }
    }

    // ---- store counts: lane holds (M = v + 8*halfSel, N = 16*nt + mlane) ----
#pragma unroll
    for (int nt = 0; nt < 3; ++nt) {
        const int col = 16 * nt + mlane;
        if (col < 36) {
#pragma unroll
            for (int v = 0; v < 8; ++v) {
                const int row = rowBase + v + 8 * halfSel;
                if (row < B) out[row * 36 + col] = counts[nt][v];
            }
        }
    }
}

extern "C" void kernel_launch(void* const* d_in, const int* in_sizes, int n_in,
                              void* d_out, int out_size, void* d_ws, size_t ws_size,
                              hipStream_t stream) {
    const float* x    = (const float*)d_in[0];
    const float* W1   = (const float*)d_in[1];
    const float* b1   = (const float*)d_in[2];
    const float* Wout = (const float*)d_in[3];
    const float* bout = (const float*)d_in[4];
    float* out = (float*)d_out;

    const int B = in_sizes[0] / 3;            // x is [B,3]
    const int blocks = (B + 127) / 128;       // 8 waves * 16 rows per block
    snn_lif_wmma_kernel<<<blocks, 256, 0, stream>>>(x, W1, b1, Wout, bout, out, B);
}